// RegionPointProcess_20083267076769
// MI455X (gfx1250) — compile-verified
//
#include <hip/hip_runtime.h>
#include <math.h>

typedef float v2f __attribute__((ext_vector_type(2)));
typedef float v8f __attribute__((ext_vector_type(8)));

#define NB 4
#define NC 256
#define NHW 16384
#define NCLS_ 19
#define NP 2048
#define NHEADS 8
#define NHD 32
#define NDH 1024
#define LN_EPS_ 1e-5f

// ---------------------------------------------------------------- helpers

__device__ __forceinline__ unsigned sortkey(float f) {
  unsigned u = __float_as_uint(f);
  return (u & 0x80000000u) ? ~u : (u | 0x80000000u);
}

__device__ __forceinline__ float gelu_tanh(float x) {
  float x3 = x * x * x;
  return 0.5f * x * (1.0f + tanhf(0.7978845608028654f * (x + 0.044715f * x3)));
}

// ---------------------------------------------------------------- entropy

__global__ __launch_bounds__(256) void entropy_kernel(const float* __restrict__ logits,
                                                      float* __restrict__ ent) {
  int id = blockIdx.x * 256 + threadIdx.x;          // < NB*NHW
  int b = id >> 14;
  int pix = id & (NHW - 1);
  const float* base = logits + (size_t)b * NCLS_ * NHW + pix;
  float m = -INFINITY;
  for (int c = 0; c < NCLS_; ++c) m = fmaxf(m, base[(size_t)c * NHW]);
  float s = 0.0f, t = 0.0f;
  for (int c = 0; c < NCLS_; ++c) {
    float l = base[(size_t)c * NHW];
    float e = expf(l - m);
    s += e;
    t += l * e;
  }
  ent[id] = m + logf(s) - t / s;   // lse - sum(p*l)
}

// ---------------------------------------------------------------- exact top-k (radix select + ordered compaction)

#define TK_PER 64  // 16384 / 256 threads

__global__ __launch_bounds__(256) void topk_kernel(const float* __restrict__ ent,
                                                   const float* __restrict__ edge,
                                                   int* __restrict__ ridx,
                                                   int* __restrict__ eidx) {
  const int src = blockIdx.x;   // 0 = entropy -> region, 1 = edge
  const int b = blockIdx.y;
  const int tid = threadIdx.x;
  const float* in = (src == 0 ? ent : edge) + (size_t)b * NHW;
  int* out = (src == 0 ? ridx : eidx) + (size_t)b * NP;

  __shared__ unsigned hist[256];
  __shared__ unsigned prefS;
  __shared__ int remS;
  __shared__ int gts[256], eqs[256];

  if (tid == 0) { prefS = 0u; remS = NP; }
  __syncthreads();

  for (int shift = 24; shift >= 0; shift -= 8) {
    hist[tid] = 0u;
    __syncthreads();
    unsigned pref = prefS;
    for (int i = 0; i < TK_PER; ++i) {
      int idx = tid * TK_PER + i;
      unsigned key = sortkey(in[idx]);
      bool match = (shift == 24) || (((key ^ pref) >> (shift + 8)) == 0u);
      if (match) atomicAdd(&hist[(key >> shift) & 255u], 1u);
    }
    __syncthreads();
    if (tid == 0) {
      int rem = remS;
      unsigned cum = 0;
      for (int d = 255; d >= 0; --d) {
        unsigned c = hist[d];
        if (cum + c >= (unsigned)rem) {
          remS = rem - (int)cum;
          prefS = prefS | ((unsigned)d << shift);
          break;
        }
        cum += c;
      }
    }
    __syncthreads();
  }

  const unsigned T = prefS;     // exact threshold key
  const int r = remS;           // #elements == T to take (lowest index first)

  int gc = 0, ec = 0;
  for (int i = 0; i < TK_PER; ++i) {
    unsigned key = sortkey(in[tid * TK_PER + i]);
    if (key > T) gc++;
    else if (key == T) ec++;
  }
  gts[tid] = gc; eqs[tid] = ec;
  __syncthreads();
  if (tid == 0) {
    int rg = 0, re = 0;
    for (int i = 0; i < 256; ++i) {
      int t1 = gts[i]; gts[i] = rg; rg += t1;
      int t2 = eqs[i]; eqs[i] = re; re += t2;
    }
  }
  __syncthreads();
  int gpos = gts[tid], epos = eqs[tid];
  const int gtTotal = NP - r;
  for (int i = 0; i < TK_PER; ++i) {
    int idx = tid * TK_PER + i;
    unsigned key = sortkey(in[idx]);
    if (key > T) {
      out[gpos++] = idx;
    } else if (key == T) {
      if (epos < r) out[gtTotal + epos] = idx;
      epos++;
    }
  }
}

// ---------------------------------------------------------------- gather + LayerNorm

__global__ __launch_bounds__(256) void gather_ln_kernel(const float* __restrict__ x,
                                                        const int* __restrict__ idx,
                                                        float* __restrict__ raw,
                                                        float* __restrict__ lnout,
                                                        const float* __restrict__ g,
                                                        const float* __restrict__ bta) {
  const int p = blockIdx.x, b = blockIdx.y, c = threadIdx.x;
  const int pix = idx[b * NP + p];
  float v = x[(size_t)b * NC * NHW + (size_t)c * NHW + pix];
  __shared__ float r1[256], r2[256];
  r1[c] = v; r2[c] = v * v;
  __syncthreads();
  for (int s = 128; s > 0; s >>= 1) {
    if (c < s) { r1[c] += r1[c + s]; r2[c] += r2[c + s]; }
    __syncthreads();
  }
  float mean = r1[0] * (1.0f / NC);
  float var = r2[0] * (1.0f / NC) - mean * mean;
  float nv = (v - mean) * rsqrtf(var + LN_EPS_);
  size_t o = (size_t)b * NP * NC + (size_t)p * NC + c;
  if (raw) raw[o] = v;
  lnout[o] = nv * g[c] + bta[c];
}

__global__ __launch_bounds__(256) void ln_rows_kernel(const float* __restrict__ in,
                                                      float* __restrict__ lnout,
                                                      const float* __restrict__ g,
                                                      const float* __restrict__ bta) {
  const int p = blockIdx.x, b = blockIdx.y, c = threadIdx.x;
  size_t o = (size_t)b * NP * NC + (size_t)p * NC + c;
  float v = in[o];
  __shared__ float r1[256], r2[256];
  r1[c] = v; r2[c] = v * v;
  __syncthreads();
  for (int s = 128; s > 0; s >>= 1) {
    if (c < s) { r1[c] += r1[c + s]; r2[c] += r2[c + s]; }
    __syncthreads();
  }
  float mean = r1[0] * (1.0f / NC);
  float var = r2[0] * (1.0f / NC) - mean * mean;
  lnout[o] = (v - mean) * rsqrtf(var + LN_EPS_) * g[c] + bta[c];
}

// ---------------------------------------------------------------- transpose (R x Cc) -> (Cc x R), dims multiples of 32

__global__ __launch_bounds__(256) void transpose_kernel(const float* __restrict__ in,
                                                        float* __restrict__ out,
                                                        int R, int Cc) {
  __shared__ float tile[32][33];
  const int tx = threadIdx.x, ty = threadIdx.y;       // block (32, 8)
  const int z = blockIdx.z;
  const size_t base = (size_t)z * R * Cc;
  const int x = blockIdx.x * 32 + tx;
  const int y0 = blockIdx.y * 32;
#pragma unroll
  for (int i = 0; i < 4; ++i) {
    int row = y0 + ty + i * 8;
    tile[ty + i * 8][tx] = in[base + (size_t)row * Cc + x];
  }
  __syncthreads();
  const int ox = y0 + tx;
  const int oy0 = blockIdx.x * 32;
#pragma unroll
  for (int i = 0; i < 4; ++i) {
    int orow = oy0 + ty + i * 8;
    out[base + (size_t)orow * R + ox] = tile[tx][ty + i * 8];
  }
}

// ---------------------------------------------------------------- GEMM: C = act(A @ W + bias) (+ residual), f32 WMMA 16x16x4
// A: [batch][M][K] row-major, WT: [N][K] (weights pre-transposed), bias: [N]
// 128 threads (4 waves, 2x2), block tile 64x64, each wave 2x2 16x16 subtiles.
// A/B tiles double-buffered in LDS, filled with GLOBAL_LOAD_ASYNC_TO_LDS_B128
// (ASYNCcnt) so the next K-chunk streams in while WMMAs consume the current one.

__global__ __launch_bounds__(128) void gemm_kernel(const float* __restrict__ A,
                                                   const float* __restrict__ WT,
                                                   const float* __restrict__ bias,
                                                   const float* __restrict__ Res,
                                                   float* __restrict__ Cm,
                                                   int M, int N, int K, int act) {
  __shared__ alignas(16) float As[2][64 * 36];   // pad 36: 16B-aligned rows, conflict-free frags
  __shared__ alignas(16) float Bs[2][64 * 36];
  const int tid = threadIdx.x;
  const int lane = tid & 31;
  const int wid = tid >> 5;
  const int hi = lane >> 4;
  const int l16 = lane & 15;
  const int wm = (wid >> 1) * 32;
  const int wn = (wid & 1) * 32;
  const int m0 = blockIdx.y * 64;
  const int n0 = blockIdx.x * 64;
  const int b = blockIdx.z;
  const size_t aOff = (size_t)b * M * K;
  const size_t cOff = (size_t)b * M * N;

  v8f a00, a01, a10, a11;
#pragma unroll
  for (int r = 0; r < 8; ++r) { a00[r] = 0.0f; a01[r] = 0.0f; a10[r] = 0.0f; a11[r] = 0.0f; }

  // async-stage one 64x32 A tile + 64x32 B tile into LDS buffer `buf`
  auto issue = [&](int buf, int k0) {
#pragma unroll
    for (int i = 0; i < 4; ++i) {
      int li = i * 128 + tid;        // 0..511 float4 chunks
      int row = li >> 3;
      int col = (li & 7) * 4;
      unsigned la = (unsigned)(size_t)&As[buf][row * 36 + col];
      unsigned long long ga =
          (unsigned long long)(size_t)(A + aOff + (size_t)(m0 + row) * K + k0 + col);
      asm volatile("global_load_async_to_lds_b128 %0, %1, off" :: "v"(la), "v"(ga) : "memory");
      unsigned lb = (unsigned)(size_t)&Bs[buf][row * 36 + col];
      unsigned long long gb =
          (unsigned long long)(size_t)(WT + (size_t)(n0 + row) * K + k0 + col);
      asm volatile("global_load_async_to_lds_b128 %0, %1, off" :: "v"(lb), "v"(gb) : "memory");
    }
  };

  const int nIter = K >> 5;
  issue(0, 0);
  for (int it = 0; it < nIter; ++it) {
    const int cur = it & 1;
    asm volatile("s_wait_asynccnt 0x0" ::: "memory");  // this wave's async LDS fills done
    __syncthreads();                                   // all waves' fills visible
    if (it + 1 < nIter) issue(1 - cur, (it + 1) << 5); // overlap next chunk with compute
    const float* as = &As[cur][0];
    const float* bs = &Bs[cur][0];
#pragma unroll
    for (int kk = 0; kk < 32; kk += 4) {
      v2f fa0 = *(const v2f*)&as[(wm + l16) * 36 + kk + hi * 2];
      v2f fa1 = *(const v2f*)&as[(wm + 16 + l16) * 36 + kk + hi * 2];
      v2f fb0 = *(const v2f*)&bs[(wn + l16) * 36 + kk + hi * 2];
      v2f fb1 = *(const v2f*)&bs[(wn + 16 + l16) * 36 + kk + hi * 2];
      a00 = __builtin_amdgcn_wmma_f32_16x16x4_f32(false, fa0, false, fb0, (short)0, a00, false, false);
      a01 = __builtin_amdgcn_wmma_f32_16x16x4_f32(false, fa0, false, fb1, (short)0, a01, false, false);
      a10 = __builtin_amdgcn_wmma_f32_16x16x4_f32(false, fa1, false, fb0, (short)0, a10, false, false);
      a11 = __builtin_amdgcn_wmma_f32_16x16x4_f32(false, fa1, false, fb1, (short)0, a11, false, false);
    }
    __syncthreads();   // everyone done reading `cur` before it is refilled
  }

  auto epi = [&](v8f& acc, int mi, int nj) {
#pragma unroll
    for (int r = 0; r < 8; ++r) {
      int row = m0 + wm + mi + hi * 8 + r;
      int col = n0 + wn + nj + l16;
      float v = acc[r] + bias[col];
      if (act == 1) v = gelu_tanh(v);
      if (Res) v += Res[cOff + (size_t)row * N + col];
      Cm[cOff + (size_t)row * N + col] = v;
    }
  };
  epi(a00, 0, 0); epi(a01, 0, 16); epi(a10, 16, 0); epi(a11, 16, 16);
}

// ---------------------------------------------------------------- attention: flash-style, 1 wave per (b, head, 16-q-row tile)
// Qb,Kb: [B][P][C], Vt: [B][C][P], O: [B][P][C]

__global__ __launch_bounds__(32) void attn_kernel(const float* __restrict__ Qb,
                                                  const float* __restrict__ Kb,
                                                  const float* __restrict__ Vt,
                                                  float* __restrict__ O) {
  const int lane = threadIdx.x;
  const int hi = lane >> 4;
  const int l16 = lane & 15;
  const int gid = blockIdx.x;
  const int qt = gid & 127;
  const int h = (gid >> 7) & 7;
  const int b = gid >> 10;
  const int q0 = qt * 16;
  const float scale = 0.17677669529663687f;   // 1/sqrt(32)

  const float* Qp = Qb + (size_t)b * NP * NC + (size_t)q0 * NC + h * NHD;
  const float* Kbase = Kb + (size_t)b * NP * NC + h * NHD;
  const float* Vbase = Vt + (size_t)b * NC * NP + (size_t)(h * NHD) * NP;

  // preload Q A-fragments for all 8 K-chunks (d = 32)
  v2f aq[8];
#pragma unroll
  for (int ch = 0; ch < 8; ++ch)
    aq[ch] = *(const v2f*)(Qp + (size_t)l16 * NC + ch * 4 + hi * 2);

  __shared__ float Pld[16 * 18];
  float m_r[8], s_r[8];
  v8f o0, o1;
#pragma unroll
  for (int r = 0; r < 8; ++r) { m_r[r] = -INFINITY; s_r[r] = 0.0f; o0[r] = 0.0f; o1[r] = 0.0f; }

  for (int j = 0; j < NP / 16; ++j) {
    const int kv0 = j * 16;
    // hide first-touch latency of the streamed K rows / V^T columns
    if (kv0 + 64 < NP) {
      __builtin_prefetch((const void*)(Kbase + (size_t)(kv0 + 64 + l16) * NC), 0, 1);
      __builtin_prefetch((const void*)(Vbase + (size_t)l16 * NP + kv0 + 64), 0, 1);
    }
    // S tile = Q(16x32) @ K_block^T -> C layout
    v8f sa;
#pragma unroll
    for (int r = 0; r < 8; ++r) sa[r] = 0.0f;
#pragma unroll
    for (int ch = 0; ch < 8; ++ch) {
      v2f bk = *(const v2f*)(Kbase + (size_t)(kv0 + l16) * NC + ch * 4 + hi * 2);
      sa = __builtin_amdgcn_wmma_f32_16x16x4_f32(false, aq[ch], false, bk, (short)0, sa, false, false);
    }
    // online softmax update, write P tile to LDS (transposed access later)
#pragma unroll
    for (int r = 0; r < 8; ++r) {
      float v = sa[r] * scale;
      float rm = v;
      for (int off = 8; off > 0; off >>= 1) rm = fmaxf(rm, __shfl_xor(rm, off, 16));
      float nm = fmaxf(m_r[r], rm);
      float resc = expf(m_r[r] - nm);
      float p = expf(v - nm);
      float ps = p;
      for (int off = 8; off > 0; off >>= 1) ps += __shfl_xor(ps, off, 16);
      s_r[r] = s_r[r] * resc + ps;
      m_r[r] = nm;
      o0[r] *= resc;
      o1[r] *= resc;
      Pld[(hi * 8 + r) * 18 + l16] = p;
    }
    __syncthreads();
    // O += P(16x16) @ V_block(16x32)
#pragma unroll
    for (int kk = 0; kk < 4; ++kk) {
      v2f ap = *(const v2f*)&Pld[l16 * 18 + kk * 4 + hi * 2];
      v2f b0 = *(const v2f*)(Vbase + (size_t)l16 * NP + kv0 + kk * 4 + hi * 2);
      o0 = __builtin_amdgcn_wmma_f32_16x16x4_f32(false, ap, false, b0, (short)0, o0, false, false);
      v2f b1 = *(const v2f*)(Vbase + (size_t)(16 + l16) * NP + kv0 + kk * 4 + hi * 2);
      o1 = __builtin_amdgcn_wmma_f32_16x16x4_f32(false, ap, false, b1, (short)0, o1, false, false);
    }
    __syncthreads();
  }

  float* Op = O + (size_t)b * NP * NC + h * NHD;
#pragma unroll
  for (int r = 0; r < 8; ++r) {
    float inv = 1.0f / s_r[r];
    size_t row = (size_t)(q0 + hi * 8 + r) * NC;
    Op[row + l16] = o0[r] * inv;
    Op[row + 16 + l16] = o1[r] * inv;
  }
}

// ---------------------------------------------------------------- scatter tokens back

__global__ __launch_bounds__(256) void scatter_kernel(const float* __restrict__ tok,
                                                      const int* __restrict__ idx,
                                                      float* __restrict__ out) {
  const int p = blockIdx.x, b = blockIdx.y, c = threadIdx.x;
  const int pix = idx[b * NP + p];
  out[(size_t)b * NC * NHW + (size_t)c * NHW + pix] =
      tok[(size_t)b * NP * NC + (size_t)p * NC + c];
}

// ---------------------------------------------------------------- launch

extern "C" void kernel_launch(void* const* d_in, const int* in_sizes, int n_in,
                              void* d_out, int out_size, void* d_ws, size_t ws_size,
                              hipStream_t stream) {
  (void)in_sizes; (void)n_in; (void)out_size; (void)ws_size;
  const float* x       = (const float*)d_in[0];
  const float* xlog    = (const float*)d_in[1];
  const float* edge    = (const float*)d_in[2];
  const float* ln_q_g  = (const float*)d_in[3];
  const float* ln_q_b  = (const float*)d_in[4];
  const float* ln_kv_g = (const float*)d_in[5];
  const float* ln_kv_b = (const float*)d_in[6];
  const float* Wq = (const float*)d_in[7];   const float* bq = (const float*)d_in[8];
  const float* Wk = (const float*)d_in[9];   const float* bk = (const float*)d_in[10];
  const float* Wv = (const float*)d_in[11];  const float* bv = (const float*)d_in[12];
  const float* Wo = (const float*)d_in[13];  const float* bo = (const float*)d_in[14];
  const float* ln2_g = (const float*)d_in[15]; const float* ln2_b = (const float*)d_in[16];
  const float* W1 = (const float*)d_in[17];  const float* b1 = (const float*)d_in[18];
  const float* W2 = (const float*)d_in[19];  const float* b2 = (const float*)d_in[20];
  float* out = (float*)d_out;

  char* ws = (char*)d_ws;
  size_t off = 0;
  auto alloc = [&](size_t bytes) -> char* {
    char* p = ws + off;
    off = (off + bytes + 255) & ~(size_t)255;
    return p;
  };
  const size_t tokBytes = (size_t)NB * NP * NC * 4;
  float* ent  = (float*)alloc((size_t)NB * NHW * 4);
  int*   ridx = (int*)alloc((size_t)NB * NP * 4);
  int*   eidx = (int*)alloc((size_t)NB * NP * 4);
  float* qraw = (float*)alloc(tokBytes);
  float* qn   = (float*)alloc(tokBytes);
  float* kn   = (float*)alloc(tokBytes);
  float* Qb   = (float*)alloc(tokBytes);
  float* Kbuf = (float*)alloc(tokBytes);
  float* Vbuf = (float*)alloc(tokBytes);
  float* Vt   = (float*)alloc(tokBytes);
  float* mlp  = (float*)alloc((size_t)NB * NP * NDH * 4);
  float* WqT  = (float*)alloc((size_t)NC * NC * 4);
  float* WkT  = (float*)alloc((size_t)NC * NC * 4);
  float* WvT  = (float*)alloc((size_t)NC * NC * 4);
  float* WoT  = (float*)alloc((size_t)NC * NC * 4);
  float* W1T  = (float*)alloc((size_t)NC * NDH * 4);
  float* W2T  = (float*)alloc((size_t)NDH * NC * 4);
  // buffer reuse (safe by dataflow order):
  float* attnB = kn;    // kn dead after K/V GEMMs
  float* h1    = qn;    // qn dead after Q GEMM
  float* hn    = Kbuf;  // Kbuf dead after attention
  float* tok   = Vbuf;  // Vbuf dead after V transpose

  // out starts as a copy of x; region columns overwritten at the end
  hipMemcpyAsync(out, x, (size_t)NB * NC * NHW * 4, hipMemcpyDeviceToDevice, stream);

  entropy_kernel<<<(NB * NHW) / 256, 256, 0, stream>>>(xlog, ent);
  topk_kernel<<<dim3(2, NB), 256, 0, stream>>>(ent, edge, ridx, eidx);

  gather_ln_kernel<<<dim3(NP, NB), 256, 0, stream>>>(x, ridx, qraw, qn, ln_q_g, ln_q_b);
  gather_ln_kernel<<<dim3(NP, NB), 256, 0, stream>>>(x, eidx, nullptr, kn, ln_kv_g, ln_kv_b);

  dim3 tb(32, 8);
  transpose_kernel<<<dim3(NC / 32, NC / 32, 1), tb, 0, stream>>>(Wq, WqT, NC, NC);
  transpose_kernel<<<dim3(NC / 32, NC / 32, 1), tb, 0, stream>>>(Wk, WkT, NC, NC);
  transpose_kernel<<<dim3(NC / 32, NC / 32, 1), tb, 0, stream>>>(Wv, WvT, NC, NC);
  transpose_kernel<<<dim3(NC / 32, NC / 32, 1), tb, 0, stream>>>(Wo, WoT, NC, NC);
  transpose_kernel<<<dim3(NDH / 32, NC / 32, 1), tb, 0, stream>>>(W1, W1T, NC, NDH);
  transpose_kernel<<<dim3(NC / 32, NDH / 32, 1), tb, 0, stream>>>(W2, W2T, NDH, NC);

  gemm_kernel<<<dim3(NC / 64, NP / 64, NB), 128, 0, stream>>>(qn, WqT, bq, nullptr, Qb, NP, NC, NC, 0);
  gemm_kernel<<<dim3(NC / 64, NP / 64, NB), 128, 0, stream>>>(kn, WkT, bk, nullptr, Kbuf, NP, NC, NC, 0);
  gemm_kernel<<<dim3(NC / 64, NP / 64, NB), 128, 0, stream>>>(kn, WvT, bv, nullptr, Vbuf, NP, NC, NC, 0);

  transpose_kernel<<<dim3(NC / 32, NP / 32, NB), tb, 0, stream>>>(Vbuf, Vt, NP, NC);

  attn_kernel<<<NB * NHEADS * (NP / 16), 32, 0, stream>>>(Qb, Kbuf, Vt, attnB);

  gemm_kernel<<<dim3(NC / 64, NP / 64, NB), 128, 0, stream>>>(attnB, WoT, bo, qraw, h1, NP, NC, NC, 0);
  ln_rows_kernel<<<dim3(NP, NB), 256, 0, stream>>>(h1, hn, ln2_g, ln2_b);
  gemm_kernel<<<dim3(NDH / 64, NP / 64, NB), 128, 0, stream>>>(hn, W1T, b1, nullptr, mlp, NP, NDH, NC, 1);
  gemm_kernel<<<dim3(NC / 64, NP / 64, NB), 128, 0, stream>>>(mlp, W2T, b2, h1, tok, NP, NC, NDH, 0);

  scatter_kernel<<<dim3(NP, NB), 256, 0, stream>>>(tok, ridx, out);
}